// NLB_9809705304994
// MI455X (gfx1250) — compile-verified
//
#include <hip/hip_runtime.h>
#include <hip/hip_bf16.h>

#define NND 100000
#define KNB 17
#define BE  4096
#define NB3 12288
#define NF  128
#define EF  128
#define SD  100
#define TD  50
#define NH  4
#define FHD 25
#define AD  406   // NF+SD+TD+EF (unpadded)
// padded K layout for the GAT GEMM (EF segment 16B-aligned for async b128):
//   cols [0,128) NF | [128,228) SD | [228,278) tenc | [278,280) pad | [280,408) EF
#define KP  408
#define EF0 280

typedef __attribute__((ext_vector_type(2))) float v2f;
typedef __attribute__((ext_vector_type(8))) float v8f;

// ---------------------------------------------------------------------------
// CDNA5 async copy: global -> LDS, 16 bytes per active lane (ASYNCcnt).
// ---------------------------------------------------------------------------
__device__ __forceinline__ void async_b128(unsigned lds_off, const void* gaddr) {
  asm volatile("global_load_async_to_lds_b128 %0, %1, off"
               :: "v"(lds_off), "v"((unsigned long long)(size_t)gaddr)
               : "memory");
}
__device__ __forceinline__ void wait_async0() {
  asm volatile("s_wait_asynccnt 0x0" ::: "memory");
}
__device__ __forceinline__ unsigned lds_off(const void* p) {
  // generic pointer to LDS: addr[31:0] is the wave-relative LDS byte offset
  return (unsigned)(size_t)p;
}

// ---------------------------------------------------------------------------
// 16x16 f32 WMMA tile accumulate: D += A(16 x 4*ksteps) * B(4*ksteps x 16)
// Lane mapping per CDNA5 ISA 7.12.2.
// ---------------------------------------------------------------------------
__device__ __forceinline__ v8f wmma_acc_f32(const float* A, int lda,
                                            const float* Bm, int ldb, int n0,
                                            int ksteps, int lane, v8f acc) {
  const int row  = lane & 15;
  const int koff = (lane >> 4) * 2;
  const int ncol = lane & 15;
  for (int kk = 0; kk < ksteps; ++kk) {
    const int kb = kk * 4 + koff;
    v2f a, b;
    a.x = A[row * lda + kb];
    a.y = A[row * lda + kb + 1];
    b.x = Bm[(kb    ) * ldb + n0 + ncol];
    b.y = Bm[(kb + 1) * ldb + n0 + ncol];
    acc = __builtin_amdgcn_wmma_f32_16x16x4_f32(false, a, false, b,
                                                (short)0, acc, false, false);
  }
  return acc;
}

// ===========================================================================
// Kernel 1: GAT — async-gathered feat tiles (double buffered), WMMA proj,
//           fused attention (leaky-relu, masked softmax over K, aggregation).
// One block = 8 batch rows -> 144 A-rows (136 ngh + 8 tgt) = 9 M-tiles.
// ===========================================================================
__global__ __launch_bounds__(256) void k_gat(
    const int* __restrict__ node_ids, const float* __restrict__ cut_time,
    const float* __restrict__ node_feats, const float* __restrict__ self_rep,
    const int* __restrict__ ngh_ids, const float* __restrict__ ngh_ts,
    const float* __restrict__ ngh_efeat, const float* __restrict__ basis_freq,
    const float* __restrict__ phase, const float* __restrict__ W_gat,
    const float* __restrict__ a_src, const float* __restrict__ a_dst,
    float* __restrict__ gat_out)
{
  extern __shared__ float lds[];
  float* Wg    = lds;                 // [408][112] (row-permuted W_gat)
  float* proj  = Wg    + KP * 112;    // [144][112]
  float* featA = proj  + 144 * 112;   // [16][408] ping
  float* featB = featA + 16 * KP;     // [16][408] pong
  float* aS    = featB + 16 * KP;     // 100
  float* aD    = aS    + 100;         // 100
  float* sS    = aD    + 100;         // [136][4] scores -> attn
  float* sT    = sS    + 136 * 4;     // [8][4]
  float* Lrts  = sT    + 32;          // 136
  float* Lct   = Lrts  + 136;         // 8
  float* Lfreq = Lct   + 8;           // 50
  float* Lph   = Lfreq + 50;          // 50
  int*   Lrid  = (int*)(Lph + 50);    // 144
  int*   Lnid  = Lrid  + 144;         // 8

  const int tid  = threadIdx.x;
  const int lane = tid & 31;
  const int wave = tid >> 5;
  const int b0   = blockIdx.x * 8;

  // W_gat into LDS with K-row permutation matching the padded feat layout
  for (int idx = tid; idx < KP * 112; idx += 256) {
    int r = idx / 112, c = idx % 112;
    float v = 0.0f;
    if (c < SD) {
      if (r < NF + SD + TD)      v = W_gat[r * SD + c];          // NF|SD|tenc
      else if (r >= EF0)         v = W_gat[(r - 2) * SD + c];    // EF (shift 2)
      // r == 278,279: pad rows -> 0
    }
    Wg[idx] = v;
  }
  if (tid < 100) { aS[tid] = a_src[tid]; aD[tid] = a_dst[tid]; }
  if (tid < 50)  { Lfreq[tid] = basis_freq[tid]; Lph[tid] = phase[tid]; }
  if (tid < 8)   { Lnid[tid] = node_ids[b0 + tid]; Lct[tid] = cut_time[b0 + tid]; }
  __syncthreads();
  if (tid < 136) {
    int bl = tid / KNB, k = tid % KNB;
    int nid = Lnid[bl];
    Lrid[tid] = ngh_ids[nid * KNB + k];
    Lrts[tid] = ngh_ts[nid * KNB + k];
  }
  if (tid >= 136 && tid < 144) Lrid[tid] = Lnid[tid - 136];
  __syncthreads();

  // ---- tile staging helpers (as lambdas over captured LDS state) ----------
  // async part: NF (32 x b128), SD (25 x b128), EF (32 x b128) per row
  auto stage_async = [&](int rbase, float* buf) {
    const unsigned base = lds_off(buf);
    for (int idx = tid; idx < 16 * 89; idx += 256) {
      int rl = idx / 89, j = idx % 89;
      int r = rbase + rl;
      unsigned ldsRow = base + (unsigned)rl * (KP * 4);
      if (r < 136) {
        int bl = r / KNB, k = r % KNB;
        int rid = Lrid[r];
        if (j < 32) {
          async_b128(ldsRow + j * 16,
                     (const char*)node_feats + (size_t)rid * (NF * 4) + j * 16);
        } else if (j < 57) {
          int jj = j - 32;
          async_b128(ldsRow + NF * 4 + jj * 16,
                     (const char*)self_rep + (size_t)rid * (SD * 4) + jj * 16);
        } else {
          int jj = j - 57;
          async_b128(ldsRow + EF0 * 4 + jj * 16,
                     (const char*)ngh_efeat +
                         ((size_t)Lnid[bl] * KNB + k) * (EF * 4) + jj * 16);
        }
      } else {
        int nid = Lnid[r - 136];
        if (j < 32) {
          async_b128(ldsRow + j * 16,
                     (const char*)node_feats + (size_t)nid * (NF * 4) + j * 16);
        } else if (j < 57) {
          int jj = j - 32;
          async_b128(ldsRow + NF * 4 + jj * 16,
                     (const char*)self_rep + (size_t)nid * (SD * 4) + jj * 16);
        }
        // EF of target rows: zeroed by stage_rest
      }
    }
  };
  // VALU part: tenc cols 228..277, pad cols 278..279, tgt-row EF zeros
  auto stage_rest = [&](int rbase, float* buf) {
    for (int idx = tid; idx < 16 * 52; idx += 256) {
      int rl = idx / 52, c = idx % 52;  // cols 228..279
      int r = rbase + rl;
      float v = 0.0f;
      if (c < TD) {
        if (r < 136) {
          int bl = r / KNB;
          v = cosf((Lct[bl] - Lrts[r]) * Lfreq[c] + Lph[c]);
        } else {
          v = cosf(Lph[c]);
        }
      }
      buf[rl * KP + (NF + SD) + c] = v;
    }
    if (rbase == 128) {  // last tile holds tgt rows 8..15: zero their EF block
      for (int idx = tid; idx < 8 * EF; idx += 256) {
        int rl = 8 + idx / EF, c = idx % EF;
        buf[rl * KP + EF0 + c] = 0.0f;
      }
    }
  };

  // ---- pipelined M-tile loop ---------------------------------------------
  stage_async(0, featA);
  stage_rest(0, featA);
  wait_async0();
  __syncthreads();

  for (int mt = 0; mt < 9; ++mt) {
    float* cur = (mt & 1) ? featB : featA;
    float* nxt = (mt & 1) ? featA : featB;
    if (mt < 8) {            // prefetch next tile while computing this one
      stage_async((mt + 1) * 16, nxt);
      stage_rest((mt + 1) * 16, nxt);
    }
    if (wave < 7) {
      v8f acc = {0.f, 0.f, 0.f, 0.f, 0.f, 0.f, 0.f, 0.f};
      acc = wmma_acc_f32(cur, KP, Wg, 112, wave * 16, KP / 4, lane, acc);
      const int rbase = mt * 16;
      const int ncol  = wave * 16 + (lane & 15);
      const int rh    = (lane >> 4) * 8;
#pragma unroll
      for (int i = 0; i < 8; ++i)
        proj[(rbase + rh + i) * 112 + ncol] = acc[i];
    }
    wait_async0();
    __syncthreads();
  }

  // ---- attention ----------------------------------------------------------
  for (int idx = tid; idx < 136 * NH; idx += 256) {
    int r = idx >> 2, h = idx & 3;
    float s = 0.0f;
#pragma unroll
    for (int f = 0; f < FHD; ++f) s += proj[r * 112 + h * FHD + f] * aS[h * FHD + f];
    sS[idx] = s;
  }
  if (tid < 32) {
    int bl = tid >> 2, h = tid & 3;
    int r = 136 + bl;
    float s = 0.0f;
#pragma unroll
    for (int f = 0; f < FHD; ++f) s += proj[r * 112 + h * FHD + f] * aD[h * FHD + f];
    sT[tid] = s;
  }
  __syncthreads();
  if (tid < 32) {  // masked leaky-relu softmax over K=17 per (b,h)
    int bl = tid >> 2, h = tid & 3;
    float ev[KNB];
    float mx = -1e30f;
    for (int k = 0; k < KNB; ++k) {
      int r = bl * KNB + k;
      float e = sS[r * 4 + h] + sT[tid];
      e = (e > 0.0f) ? e : 0.2f * e;
      if (Lrid[r] == 0) e = -1e9f;
      ev[k] = e;
      mx = fmaxf(mx, e);
    }
    float sum = 0.0f;
    for (int k = 0; k < KNB; ++k) { ev[k] = __expf(ev[k] - mx); sum += ev[k]; }
    float inv = 1.0f / sum;
    for (int k = 0; k < KNB; ++k) {
      int r = bl * KNB + k;
      sS[r * 4 + h] = (Lrid[r] == 0) ? 0.0f : ev[k] * inv;
    }
  }
  __syncthreads();
  for (int idx = tid; idx < 8 * SD; idx += 256) {
    int bl = idx / SD, f = idx % SD;
    int h = f / FHD;
    float s = 0.0f;
#pragma unroll
    for (int k = 0; k < KNB; ++k) {
      int r = bl * KNB + k;
      s += sS[r * 4 + h] * proj[r * 112 + f];
    }
    gat_out[(size_t)(b0 + bl) * SD + f] = s;
  }
}

// ===========================================================================
// Kernel 2a: x = [h0, tenc_p, prev_ef](278) @ W_selfrep(278x100)
// ===========================================================================
__global__ __launch_bounds__(256) void k_selfproj(
    const int* __restrict__ node_ids, const float* __restrict__ cut_time,
    const float* __restrict__ self_rep, const float* __restrict__ ngh_ts,
    const float* __restrict__ ngh_efeat, const float* __restrict__ basis_freq,
    const float* __restrict__ phase, const float* __restrict__ W_selfrep,
    float* __restrict__ xout)
{
  extern __shared__ float lds[];
  float* Ws = lds;             // [280][112]
  float* As = Ws + 280 * 112;  // [16][280]
  const int tid = threadIdx.x, lane = tid & 31, wave = tid >> 5;
  const int b0 = blockIdx.x * 16;

  for (int idx = tid; idx < 280 * 112; idx += 256) {
    int r = idx / 112, c = idx % 112;
    Ws[idx] = (r < SD + TD + EF && c < SD) ? W_selfrep[r * SD + c] : 0.0f;
  }
  for (int idx = tid; idx < 16 * 280; idx += 256) {
    int rl = idx / 280, c = idx % 280;
    int b = b0 + rl;
    int nid = node_ids[b];
    float v = 0.0f;
    if (c < SD)            v = self_rep[(size_t)nid * SD + c];
    else if (c < SD + TD) {
      int j = c - SD;
      float dt = cut_time[b] - ngh_ts[nid * KNB];
      v = cosf(dt * basis_freq[j] + phase[j]);
    } else if (c < SD + TD + EF) {
      v = ngh_efeat[(size_t)nid * (KNB * EF) + (c - SD - TD)];
    }
    As[idx] = v;
  }
  __syncthreads();
  if (wave < 7) {
    v8f acc = {0.f, 0.f, 0.f, 0.f, 0.f, 0.f, 0.f, 0.f};
    acc = wmma_acc_f32(As, 280, Ws, 112, wave * 16, 70, lane, acc);
    int n = wave * 16 + (lane & 15);
    int rh = (lane >> 4) * 8;
    if (n < SD) {
#pragma unroll
      for (int i = 0; i < 8; ++i)
        xout[(size_t)(b0 + rh + i) * SD + n] = acc[i];
    }
  }
}

// ===========================================================================
// Kernel 2b: GRU — gi = x@W_ih, gh = h0@W_hh (same LDS slab, two passes),
//            fused gates -> h_new
// ===========================================================================
__global__ __launch_bounds__(256) void k_gru(
    const int* __restrict__ node_ids, const float* __restrict__ self_rep,
    const float* __restrict__ W_ih, const float* __restrict__ W_hh,
    const float* __restrict__ b_ih, const float* __restrict__ b_hh,
    const float* __restrict__ xin, float* __restrict__ hnew)
{
  extern __shared__ float lds[];
  float* Wl  = lds;             // [104][304]
  float* xs  = Wl  + 104 * 304; // [16][104]
  float* h0s = xs  + 16 * 104;  // [16][104]
  float* gi  = h0s + 16 * 104;  // [16][304]
  float* gh  = gi  + 16 * 304;  // [16][304]
  const int tid = threadIdx.x, lane = tid & 31, wave = tid >> 5;
  const int b0 = blockIdx.x * 16;

  for (int idx = tid; idx < 16 * 104; idx += 256) {
    int rl = idx / 104, c = idx % 104;
    int b = b0 + rl;
    xs[idx]  = (c < SD) ? xin[(size_t)b * SD + c] : 0.0f;
    h0s[idx] = (c < SD) ? self_rep[(size_t)node_ids[b] * SD + c] : 0.0f;
  }
  for (int idx = tid; idx < 104 * 304; idx += 256) {
    int r = idx / 304, c = idx % 304;
    Wl[idx] = (r < SD && c < 3 * SD) ? W_ih[r * (3 * SD) + c] : 0.0f;
  }
  __syncthreads();
  for (int nt = wave; nt < 19; nt += 8) {
    v8f acc = {0.f, 0.f, 0.f, 0.f, 0.f, 0.f, 0.f, 0.f};
    acc = wmma_acc_f32(xs, 104, Wl, 304, nt * 16, 26, lane, acc);
    int n = nt * 16 + (lane & 15);
    int rh = (lane >> 4) * 8;
#pragma unroll
    for (int i = 0; i < 8; ++i) gi[(rh + i) * 304 + n] = acc[i];
  }
  __syncthreads();
  for (int idx = tid; idx < 104 * 304; idx += 256) {
    int r = idx / 304, c = idx % 304;
    Wl[idx] = (r < SD && c < 3 * SD) ? W_hh[r * (3 * SD) + c] : 0.0f;
  }
  __syncthreads();
  for (int nt = wave; nt < 19; nt += 8) {
    v8f acc = {0.f, 0.f, 0.f, 0.f, 0.f, 0.f, 0.f, 0.f};
    acc = wmma_acc_f32(h0s, 104, Wl, 304, nt * 16, 26, lane, acc);
    int n = nt * 16 + (lane & 15);
    int rh = (lane >> 4) * 8;
#pragma unroll
    for (int i = 0; i < 8; ++i) gh[(rh + i) * 304 + n] = acc[i];
  }
  __syncthreads();
  for (int idx = tid; idx < 16 * SD; idx += 256) {
    int rl = idx / SD, f = idx % SD;
    float ir  = gi[rl * 304 + f]           + b_ih[f];
    float iz  = gi[rl * 304 + SD + f]      + b_ih[SD + f];
    float in_ = gi[rl * 304 + 2 * SD + f]  + b_ih[2 * SD + f];
    float hr  = gh[rl * 304 + f]           + b_hh[f];
    float hz  = gh[rl * 304 + SD + f]      + b_hh[SD + f];
    float hn  = gh[rl * 304 + 2 * SD + f]  + b_hh[2 * SD + f];
    float r = 1.0f / (1.0f + __expf(-(ir + hr)));
    float z = 1.0f / (1.0f + __expf(-(iz + hz)));
    float n = tanhf(in_ + r * hn);
    float h0 = h0s[rl * 104 + f];
    hnew[(size_t)(b0 + rl) * SD + f] = (1.0f - z) * n + z * h0;
  }
}

// ===========================================================================
// Kernel 2c: merged = relu([h_new, gat_out]@fc1 + b1); emb = merged@fc2 + b2
// ===========================================================================
__global__ __launch_bounds__(256) void k_merge(
    const float* __restrict__ hnew, const float* __restrict__ gat,
    const float* __restrict__ fc1_w, const float* __restrict__ fc1_b,
    const float* __restrict__ fc2_w, const float* __restrict__ fc2_b,
    float* __restrict__ emb)
{
  extern __shared__ float lds[];
  float* W1 = lds;             // [208][112]
  float* W2 = W1 + 208 * 112;  // [104][112]
  float* As = W2 + 104 * 112;  // [16][208]
  float* Ms = As + 16 * 208;   // [16][104]
  const int tid = threadIdx.x, lane = tid & 31, wave = tid >> 5;
  const int b0 = blockIdx.x * 16;

  for (int idx = tid; idx < 208 * 112; idx += 256) {
    int r = idx / 112, c = idx % 112;
    W1[idx] = (r < 2 * SD && c < SD) ? fc1_w[r * SD + c] : 0.0f;
  }
  for (int idx = tid; idx < 104 * 112; idx += 256) {
    int r = idx / 112, c = idx % 112;
    W2[idx] = (r < SD && c < SD) ? fc2_w[r * SD + c] : 0.0f;
  }
  for (int idx = tid; idx < 16 * 208; idx += 256) {
    int rl = idx / 208, c = idx % 208;
    int b = b0 + rl;
    float v = 0.0f;
    if (c < SD)          v = hnew[(size_t)b * SD + c];
    else if (c < 2 * SD) v = gat[(size_t)b * SD + (c - SD)];
    As[idx] = v;
  }
  __syncthreads();
  if (wave < 7) {
    v8f acc = {0.f, 0.f, 0.f, 0.f, 0.f, 0.f, 0.f, 0.f};
    acc = wmma_acc_f32(As, 208, W1, 112, wave * 16, 52, lane, acc);
    int n = wave * 16 + (lane & 15);
    int rh = (lane >> 4) * 8;
    if (n < 104) {
#pragma unroll
      for (int i = 0; i < 8; ++i) {
        float v = (n < SD) ? fmaxf(acc[i] + fc1_b[n], 0.0f) : 0.0f;
        Ms[(rh + i) * 104 + n] = v;
      }
    }
  }
  __syncthreads();
  if (wave < 7) {
    v8f acc = {0.f, 0.f, 0.f, 0.f, 0.f, 0.f, 0.f, 0.f};
    acc = wmma_acc_f32(Ms, 104, W2, 112, wave * 16, 26, lane, acc);
    int n = wave * 16 + (lane & 15);
    int rh = (lane >> 4) * 8;
    if (n < SD) {
#pragma unroll
      for (int i = 0; i < 8; ++i)
        emb[(size_t)(b0 + rh + i) * SD + n] = acc[i] + fc2_b[n];
    }
  }
}

// ===========================================================================
// Kernel 3: h_src/h_pos/h_neg projections + fused relu-dot with out_w
// ===========================================================================
__global__ __launch_bounds__(256) void k_head(
    const float* __restrict__ emb,
    const float* __restrict__ src_w, const float* __restrict__ src_b,
    const float* __restrict__ dst_w, const float* __restrict__ dst_b,
    const float* __restrict__ out_w, const float* __restrict__ out_b,
    float* __restrict__ out)
{
  extern __shared__ float lds[];
  float* Wsrc = lds;              // [104][112]
  float* Wdst = Wsrc + 104 * 112; // [104][112]
  float* As   = Wdst + 104 * 112; // [16][104]
  float* Hs   = As   + 16 * 104;  // h_src [16][104]
  float* Hp   = Hs   + 16 * 104;  // h_pos
  float* Hn   = Hp   + 16 * 104;  // h_neg
  const int tid = threadIdx.x, lane = tid & 31, wave = tid >> 5;
  const int e0 = blockIdx.x * 16;

  for (int idx = tid; idx < 104 * 112; idx += 256) {
    int r = idx / 112, c = idx % 112;
    bool ok = (r < SD && c < SD);
    Wsrc[idx] = ok ? src_w[r * SD + c] : 0.0f;
    Wdst[idx] = ok ? dst_w[r * SD + c] : 0.0f;
  }
  for (int p = 0; p < 3; ++p) {
    __syncthreads();
    for (int idx = tid; idx < 16 * 104; idx += 256) {
      int rl = idx / 104, c = idx % 104;
      int row = p * BE + e0 + rl;
      As[idx] = (c < SD) ? emb[(size_t)row * SD + c] : 0.0f;
    }
    __syncthreads();
    if (wave < 7) {
      const float* W  = (p == 0) ? Wsrc : Wdst;
      const float* bb = (p == 0) ? src_b : dst_b;
      float* R = (p == 0) ? Hs : ((p == 1) ? Hp : Hn);
      v8f acc = {0.f, 0.f, 0.f, 0.f, 0.f, 0.f, 0.f, 0.f};
      acc = wmma_acc_f32(As, 104, W, 112, wave * 16, 26, lane, acc);
      int n = wave * 16 + (lane & 15);
      int rh = (lane >> 4) * 8;
      if (n < SD) {
#pragma unroll
        for (int i = 0; i < 8; ++i)
          R[(rh + i) * 104 + n] = acc[i] + bb[n];
      }
    }
  }
  __syncthreads();
  if (tid < 32) {
    int e = tid & 15;
    int isneg = tid >> 4;
    const float* Hx = isneg ? Hn : Hp;
    float s = 0.0f;
#pragma unroll 4
    for (int f = 0; f < SD; ++f) {
      float v = fmaxf(Hs[e * 104 + f] + Hx[e * 104 + f], 0.0f);
      s += v * out_w[f];
    }
    out[isneg * BE + e0 + e] = s + out_b[0];
  }
}

// ===========================================================================
// Host launcher
// ===========================================================================
extern "C" void kernel_launch(void* const* d_in, const int* in_sizes, int n_in,
                              void* d_out, int out_size, void* d_ws, size_t ws_size,
                              hipStream_t stream) {
  const int*   node_ids   = (const int*)  d_in[0];
  const float* cut_time   = (const float*)d_in[1];
  const float* node_feats = (const float*)d_in[2];
  const float* self_rep   = (const float*)d_in[3];
  const int*   ngh_ids    = (const int*)  d_in[4];
  const float* ngh_ts     = (const float*)d_in[5];
  const float* ngh_efeat  = (const float*)d_in[6];
  const float* basis_freq = (const float*)d_in[7];
  const float* phase      = (const float*)d_in[8];
  const float* W_gat      = (const float*)d_in[9];
  const float* a_src      = (const float*)d_in[10];
  const float* a_dst      = (const float*)d_in[11];
  const float* W_selfrep  = (const float*)d_in[12];
  const float* W_ih       = (const float*)d_in[13];
  const float* W_hh       = (const float*)d_in[14];
  const float* b_ih       = (const float*)d_in[15];
  const float* b_hh       = (const float*)d_in[16];
  const float* fc1_w      = (const float*)d_in[17];
  const float* fc1_b      = (const float*)d_in[18];
  const float* fc2_w      = (const float*)d_in[19];
  const float* fc2_b      = (const float*)d_in[20];
  const float* src_w      = (const float*)d_in[21];
  const float* src_b      = (const float*)d_in[22];
  const float* dst_w      = (const float*)d_in[23];
  const float* dst_b      = (const float*)d_in[24];
  const float* out_w      = (const float*)d_in[25];
  const float* out_b      = (const float*)d_in[26];

  float* ws   = (float*)d_ws;
  float* gat  = ws;                          // NB3*SD
  float* xbuf = gat  + (size_t)NB3 * SD;     // NB3*SD
  float* hnew = xbuf + (size_t)NB3 * SD;     // NB3*SD
  float* emb  = hnew + (size_t)NB3 * SD;     // NB3*SD

  const size_t LDS_GAT   = (size_t)(KP*112 + 144*112 + 16*KP*2 + 100 + 100 +
                                    136*4 + 32 + 136 + 8 + 50 + 50 + 144 + 8) * 4;
  const size_t LDS_SELF  = (size_t)(280*112 + 16*280) * 4;
  const size_t LDS_GRU   = (size_t)(104*304 + 16*104*2 + 16*304*2) * 4;
  const size_t LDS_MERGE = (size_t)(208*112 + 104*112 + 16*208 + 16*104) * 4;
  const size_t LDS_HEAD  = (size_t)(104*112*2 + 16*104*4) * 4;

  (void)hipFuncSetAttribute(reinterpret_cast<const void*>(k_gat),
        hipFuncAttributeMaxDynamicSharedMemorySize, (int)LDS_GAT);
  (void)hipFuncSetAttribute(reinterpret_cast<const void*>(k_selfproj),
        hipFuncAttributeMaxDynamicSharedMemorySize, (int)LDS_SELF);
  (void)hipFuncSetAttribute(reinterpret_cast<const void*>(k_gru),
        hipFuncAttributeMaxDynamicSharedMemorySize, (int)LDS_GRU);
  (void)hipFuncSetAttribute(reinterpret_cast<const void*>(k_merge),
        hipFuncAttributeMaxDynamicSharedMemorySize, (int)LDS_MERGE);
  (void)hipFuncSetAttribute(reinterpret_cast<const void*>(k_head),
        hipFuncAttributeMaxDynamicSharedMemorySize, (int)LDS_HEAD);

  k_gat<<<NB3 / 8, 256, LDS_GAT, stream>>>(
      node_ids, cut_time, node_feats, self_rep, ngh_ids, ngh_ts, ngh_efeat,
      basis_freq, phase, W_gat, a_src, a_dst, gat);

  k_selfproj<<<NB3 / 16, 256, LDS_SELF, stream>>>(
      node_ids, cut_time, self_rep, ngh_ts, ngh_efeat, basis_freq, phase,
      W_selfrep, xbuf);

  k_gru<<<NB3 / 16, 256, LDS_GRU, stream>>>(
      node_ids, self_rep, W_ih, W_hh, b_ih, b_hh, xbuf, hnew);

  k_merge<<<NB3 / 16, 256, LDS_MERGE, stream>>>(
      hnew, gat, fc1_w, fc1_b, fc2_w, fc2_b, emb);

  k_head<<<BE / 16, 256, LDS_HEAD, stream>>>(
      emb, src_w, src_b, dst_w, dst_b, out_w, out_b, (float*)d_out);
}